// GNN_Redisual_Feature_Extractor_77189152243917
// MI455X (gfx1250) — compile-verified
//
#include <hip/hip_runtime.h>

// ---------------------------------------------------------------------------
// GNN residual feature extractor for MI455X (gfx1250, wave32).
//
// Strategy:
//  * GEMMs (x@W per conv) use V_WMMA_F32_16X16X4_F32 — full fp32 matrix path,
//    one wave per 16x16 output tile, K stepped by 4 (8/16/32 WMMA per tile).
//  * Edge aggregation = float4 gathers of h[src] + f32 atomics into out[dst].
//    All activation arrays (<=51MB) fit in the 192MB L2, so atomics/gathers
//    are L2-resident; this is the bandwidth-bound part of the kernel.
//  * Degree (with self-loop weight 1) and dinv = rsqrt(deg) computed once.
// ---------------------------------------------------------------------------

typedef float v2f __attribute__((ext_vector_type(2)));
typedef float v8f __attribute__((ext_vector_type(8)));

// ---------------- small elementwise / setup kernels ------------------------

__global__ void embed_kernel(const float* __restrict__ feat,
                             const int* __restrict__ idx,
                             const float* __restrict__ W,
                             const float* __restrict__ b,
                             float* __restrict__ X, int fin, int count) {
  int gid = blockIdx.x * blockDim.x + threadIdx.x;
  if (gid >= count * 32) return;
  int n = gid >> 5;
  int f = gid & 31;
  const float* fr = feat + (size_t)n * fin;
  float acc = b[f];
  for (int k = 0; k < fin; ++k) acc += fr[k] * W[k * 32 + f];
  X[(size_t)idx[n] * 32 + f] = acc;
}

__global__ void init_ones_kernel(float* __restrict__ p, int n) {
  int gid = blockIdx.x * blockDim.x + threadIdx.x;
  if (gid < n) p[gid] = 1.0f;   // self-loop weight
}

__global__ void deg_scatter_kernel(const int* __restrict__ dst,
                                   float* __restrict__ deg, int e) {
  int gid = blockIdx.x * blockDim.x + threadIdx.x;
  if (gid < e) atomicAdd(&deg[dst[gid]], 1.0f);
}

__global__ void rsqrt_kernel(float* __restrict__ p, int n) {
  int gid = blockIdx.x * blockDim.x + threadIdx.x;
  if (gid < n) p[gid] = __frsqrt_rn(p[gid]);
}

// ---------------- fp32 WMMA GEMM: H[nrows,FOUT] = A[nrows,FIN] @ W ---------
// One wave per 16x16 tile of H. nrows must be a multiple of 16 (100000 is).
// Layouts per CDNA5 ISA 7.12.2:
//   A 16x4 f32: lanes 0-15 -> K=k0+{0,1}, lanes 16-31 -> K=k0+{2,3}, M=lane&15
//   B 4x16 f32: mirrored (N=lane&15)
//   C/D 16x16:  VGPR r -> M=r (lanes 0-15) / M=r+8 (lanes 16-31), N=lane&15

template <int FIN, int FOUT>
__global__ void gemm_f32_wmma(const float* __restrict__ A,
                              const float* __restrict__ W,
                              float* __restrict__ H, int nrows) {
  const int lane = threadIdx.x & 31;
  const int wave = (blockIdx.x * blockDim.x + threadIdx.x) >> 5;
  const int tilesN = FOUT / 16;
  const int tilesM = nrows / 16;
  if (wave >= tilesM * tilesN) return;
  const int tm = wave / tilesN;
  const int tn = wave % tilesN;

  const int ml = lane & 15;          // M index (A) / N index (B, store)
  const int kh = (lane >> 4) << 1;   // 0 for lanes 0-15, 2 for lanes 16-31
  const float* arow = A + (size_t)(tm * 16 + ml) * FIN;
  const int col = tn * 16 + ml;

  v8f c = {};
#pragma unroll
  for (int k0 = 0; k0 < FIN; k0 += 4) {
    v2f a, b;
    a.x = arow[k0 + kh];
    a.y = arow[k0 + kh + 1];
    b.x = W[(size_t)(k0 + kh) * FOUT + col];
    b.y = W[(size_t)(k0 + kh + 1) * FOUT + col];
    c = __builtin_amdgcn_wmma_f32_16x16x4_f32(
        /*neg_a=*/false, a, /*neg_b=*/false, b,
        /*c_mod=*/(short)0, c, /*reuse_a=*/false, /*reuse_b=*/false);
  }

  const int moff = (lane >> 4) << 3;  // lanes 16-31 hold rows M=8..15
  float* hrow = H + (size_t)(tm * 16) * FOUT + col;
#pragma unroll
  for (int r = 0; r < 8; ++r) hrow[(size_t)(r + moff) * FOUT] = c[r];
}

// ---------------- GCN pieces ----------------------------------------------

// OUT[n,f] = H[n,f] * dinv[n]^2 + b[f]   (self-loop term + bias)
__global__ void self_bias_kernel(const float* __restrict__ H,
                                 const float* __restrict__ dinv,
                                 const float* __restrict__ b,
                                 float* __restrict__ OUT, int F,
                                 long long total) {
  long long gid = (long long)blockIdx.x * blockDim.x + threadIdx.x;
  if (gid >= total) return;
  int n = (int)(gid / F);
  int f = (int)(gid % F);
  float di = dinv[n];
  OUT[gid] = H[gid] * di * di + b[f];
}

// OUT[dst] += H[src] * dinv[src]*dinv[dst], float4 per thread + f32 atomics
__global__ void edge_scatter_kernel(const int* __restrict__ src,
                                    const int* __restrict__ dst,
                                    const float* __restrict__ dinv,
                                    const float* __restrict__ H,
                                    float* __restrict__ OUT, int F4,
                                    long long total) {
  long long gid = (long long)blockIdx.x * blockDim.x + threadIdx.x;
  if (gid >= total) return;
  int e = (int)(gid / F4);
  int c = (int)(gid % F4);
  int s = src[e];
  int d = dst[e];
  float w = dinv[s] * dinv[d];
  const float4 hv = ((const float4*)(H + (size_t)s * F4 * 4))[c];
  float* o = OUT + (size_t)d * F4 * 4 + (size_t)c * 4;
  atomicAdd(o + 0, hv.x * w);
  atomicAdd(o + 1, hv.y * w);
  atomicAdd(o + 2, hv.z * w);
  atomicAdd(o + 3, hv.w * w);
}

__global__ void relu_copy_kernel(const float* __restrict__ in,
                                 float* __restrict__ out, long long total) {
  long long gid = (long long)blockIdx.x * blockDim.x + threadIdx.x;
  if (gid < total) out[gid] = fmaxf(in[gid], 0.0f);
}

__global__ void relu_inplace_kernel(float* __restrict__ p, long long total) {
  long long gid = (long long)blockIdx.x * blockDim.x + threadIdx.x;
  if (gid < total) p[gid] = fmaxf(p[gid], 0.0f);
}

// p = relu(p) + res   (residual add of pre-relu conv1 output x1)
__global__ void relu_residual_kernel(float* __restrict__ p,
                                     const float* __restrict__ res,
                                     long long total) {
  long long gid = (long long)blockIdx.x * blockDim.x + threadIdx.x;
  if (gid < total) p[gid] = fmaxf(p[gid], 0.0f) + res[gid];
}

// ---------------- pooling --------------------------------------------------

__global__ void zero_kernel(float* __restrict__ p, long long total) {
  long long gid = (long long)blockIdx.x * blockDim.x + threadIdx.x;
  if (gid < total) p[gid] = 0.0f;
}

__global__ void pool_accum_kernel(const float* __restrict__ h,
                                  const int* __restrict__ batch,
                                  float* __restrict__ sums,
                                  float* __restrict__ cnt, int n, int F) {
  long long gid = (long long)blockIdx.x * blockDim.x + threadIdx.x;
  if (gid >= (long long)n * F) return;
  int node = (int)(gid / F);
  int f = (int)(gid % F);
  int g = batch[node];
  atomicAdd(&sums[(size_t)g * F + f], h[gid]);
  if (f == 0) atomicAdd(&cnt[g], 1.0f);
}

__global__ void pool_div_kernel(const float* __restrict__ sums,
                                const float* __restrict__ cnt,
                                float* __restrict__ out, int F,
                                long long total) {
  long long gid = (long long)blockIdx.x * blockDim.x + threadIdx.x;
  if (gid >= total) return;
  int g = (int)(gid / F);
  out[gid] = sums[gid] / cnt[g];
}

// ---------------- host driver ----------------------------------------------

static inline unsigned grid1(long long total, int bs) {
  return (unsigned)((total + bs - 1) / bs);
}

extern "C" void kernel_launch(void* const* d_in, const int* in_sizes, int n_in,
                              void* d_out, int out_size, void* d_ws,
                              size_t ws_size, hipStream_t stream) {
  // inputs per setup_inputs() order
  const float* ev_f  = (const float*)d_in[0];
  const float* cs_f  = (const float*)d_in[1];
  const float* tr_f  = (const float*)d_in[2];
  const float* env_f = (const float*)d_in[3];
  const int*   edge  = (const int*)d_in[4];
  const int*   ev_i  = (const int*)d_in[5];
  const int*   cs_i  = (const int*)d_in[6];
  const int*   tr_i  = (const int*)d_in[7];
  const int*   env_i = (const int*)d_in[8];
  const int*   batch = (const int*)d_in[9];
  const float* W_ev = (const float*)d_in[10]; const float* b_ev = (const float*)d_in[11];
  const float* W_cs = (const float*)d_in[12]; const float* b_cs = (const float*)d_in[13];
  const float* W_tr = (const float*)d_in[14]; const float* b_tr = (const float*)d_in[15];
  const float* W_env= (const float*)d_in[16]; const float* b_env= (const float*)d_in[17];
  const float* W1 = (const float*)d_in[18]; const float* b1 = (const float*)d_in[19];
  const float* W2 = (const float*)d_in[20]; const float* b2 = (const float*)d_in[21];
  const float* W3 = (const float*)d_in[22]; const float* b3 = (const float*)d_in[23];

  const int E = in_sizes[4] / 2;
  const int* src = edge;
  const int* dst = edge + E;
  const int N = in_sizes[9];

  const int n_ev  = in_sizes[5], f_ev  = in_sizes[0] / n_ev;
  const int n_cs  = in_sizes[6], f_cs  = in_sizes[1] / n_cs;
  const int n_tr  = in_sizes[7], f_tr  = in_sizes[2] / n_tr;
  const int n_env = in_sizes[8], f_env = in_sizes[3] / n_env;
  const int n_graphs = out_size / 64;

  // workspace layout (floats)
  float* ws = (float*)d_ws;
  size_t off = 0;
  float* X  = ws + off; off += (size_t)N * 32;   // embedded input
  float* X1 = ws + off; off += (size_t)N * 64;   // conv1 output (pre-relu), residual src
  float* R  = ws + off; off += (size_t)N * 64;   // relu buffer / conv3 output
  float* G  = ws + off; off += (size_t)N * 128;  // GEMM output h = x@W
  float* O2 = ws + off; off += (size_t)N * 128;  // conv2 output
  float* DI = ws + off; off += (size_t)N;        // dinv = rsqrt(deg)
  float* SUM= ws + off; off += (size_t)n_graphs * 64;
  float* CNT= ws + off; off += (size_t)n_graphs;
  (void)ws_size; (void)n_in;

  const int BS = 256;

  // 1) type embeddings -> X[N,32]
  embed_kernel<<<grid1((long long)n_ev  * 32, BS), BS, 0, stream>>>(ev_f,  ev_i,  W_ev,  b_ev,  X, f_ev,  n_ev);
  embed_kernel<<<grid1((long long)n_cs  * 32, BS), BS, 0, stream>>>(cs_f,  cs_i,  W_cs,  b_cs,  X, f_cs,  n_cs);
  embed_kernel<<<grid1((long long)n_tr  * 32, BS), BS, 0, stream>>>(tr_f,  tr_i,  W_tr,  b_tr,  X, f_tr,  n_tr);
  embed_kernel<<<grid1((long long)n_env * 32, BS), BS, 0, stream>>>(env_f, env_i, W_env, b_env, X, f_env, n_env);

  // 2) degrees (with self-loop) and dinv
  init_ones_kernel<<<grid1(N, BS), BS, 0, stream>>>(DI, N);
  deg_scatter_kernel<<<grid1(E, BS), BS, 0, stream>>>(dst, DI, E);
  rsqrt_kernel<<<grid1(N, BS), BS, 0, stream>>>(DI, N);

  // 3) conv1: x1 = GCN(X, W1, b1)   [N,64]
  {
    long long waves = (long long)(N / 16) * (64 / 16);
    gemm_f32_wmma<32, 64><<<grid1(waves * 32, 128), 128, 0, stream>>>(X, W1, G, N);
    self_bias_kernel<<<grid1((long long)N * 64, BS), BS, 0, stream>>>(G, DI, b1, X1, 64, (long long)N * 64);
    edge_scatter_kernel<<<grid1((long long)E * 16, BS), BS, 0, stream>>>(src, dst, DI, G, X1, 16, (long long)E * 16);
    relu_copy_kernel<<<grid1((long long)N * 64, BS), BS, 0, stream>>>(X1, R, (long long)N * 64);
  }

  // 4) conv2: O2 = relu(GCN(R, W2, b2))   [N,128]
  {
    long long waves = (long long)(N / 16) * (128 / 16);
    gemm_f32_wmma<64, 128><<<grid1(waves * 32, 128), 128, 0, stream>>>(R, W2, G, N);
    self_bias_kernel<<<grid1((long long)N * 128, BS), BS, 0, stream>>>(G, DI, b2, O2, 128, (long long)N * 128);
    edge_scatter_kernel<<<grid1((long long)E * 32, BS), BS, 0, stream>>>(src, dst, DI, G, O2, 32, (long long)E * 32);
    relu_inplace_kernel<<<grid1((long long)N * 128, BS), BS, 0, stream>>>(O2, (long long)N * 128);
  }

  // 5) conv3: R = relu(GCN(O2, W3, b3)) + X1   [N,64]
  {
    long long waves = (long long)(N / 16) * (64 / 16);
    gemm_f32_wmma<128, 64><<<grid1(waves * 32, 128), 128, 0, stream>>>(O2, W3, G, N);
    self_bias_kernel<<<grid1((long long)N * 64, BS), BS, 0, stream>>>(G, DI, b3, R, 64, (long long)N * 64);
    edge_scatter_kernel<<<grid1((long long)E * 16, BS), BS, 0, stream>>>(src, dst, DI, G, R, 16, (long long)E * 16);
    relu_residual_kernel<<<grid1((long long)N * 64, BS), BS, 0, stream>>>(R, X1, (long long)N * 64);
  }

  // 6) global mean pool -> d_out[n_graphs, 64]
  zero_kernel<<<grid1((long long)n_graphs * 64, BS), BS, 0, stream>>>(SUM, (long long)n_graphs * 64);
  zero_kernel<<<grid1(n_graphs, BS), BS, 0, stream>>>(CNT, n_graphs);
  pool_accum_kernel<<<grid1((long long)N * 64, BS), BS, 0, stream>>>(R, batch, SUM, CNT, N, 64);
  pool_div_kernel<<<grid1((long long)n_graphs * 64, BS), BS, 0, stream>>>(SUM, CNT, (float*)d_out, 64, (long long)n_graphs * 64);
}